// LIFCell_30236569764343
// MI455X (gfx1250) — compile-verified
//
#include <hip/hip_runtime.h>

// LIF scan over T=128 timesteps, N neurons, fully parallel in n.
// Memory-bound streaming: ~537 MB traffic -> ~23 us floor at 23.3 TB/s.
// No matrix structure (AI ~ 0.3 FLOP/byte), so the CDNA5-optimal shape is
// wave32 NT b128 streaming with deep load pipelining, not WMMA.

typedef float v4f __attribute__((ext_vector_type(4)));

#define LIF_T      128
#define LIF_BETA   0.95f
#define LIF_GAMMA  0.95f
#define LIF_THRESH 1.0f

__device__ __forceinline__ void lif_step(v4f x, v4f& v, v4f& s)
{
    v4f vn, sn;
#pragma unroll
    for (int k = 0; k < 4; ++k) {
        // v' = beta*v + I_t - gamma*s
        float vv = __builtin_fmaf(LIF_BETA, v[k], x[k]);
        vv = __builtin_fmaf(-LIF_GAMMA, s[k], vv);
        // s' = (v' > th);  v'' = (v' > th) ? 0 : v'
        sn[k] = (vv > LIF_THRESH) ? 1.0f : 0.0f;
        vn[k] = (vv > LIF_THRESH) ? 0.0f : vv;
    }
    v = vn;
    s = sn;
}

// ---------------------------------------------------------------------------
// Reference-shape kernel (N = 262144, n4 = 65536). Defined FIRST so the
// disasm snippet shows its inner loop. Compile-time stride: t*N4*16B folds
// into the signed 24-bit vmem ioffset (unroll-8 group spans 0..7 MB), so the
// compiler keeps one base pointer per array, bumps it once per 8-step trip,
// and emits immediate-offset global_load_b128/global_store_b128 th:NT.
// ---------------------------------------------------------------------------
__global__ __launch_bounds__(256, 1) void lif_scan_v4_ref(
    const v4f* __restrict__ I,
    v4f* __restrict__ spk,
    v4f* __restrict__ vm,
    v4f* __restrict__ spk2)
{
    constexpr int N4 = 262144 / 4;   // 65536 float4 per row; row = 1 MB

    const int i = blockIdx.x * blockDim.x + threadIdx.x;

    const v4f* ip  = I    + i;
    v4f*       sp  = spk  + i;
    v4f*       vp  = vm   + i;
    v4f*       sp2 = spk2 + i;

    v4f v = {0.0f, 0.0f, 0.0f, 0.0f};
    v4f s = {0.0f, 0.0f, 0.0f, 0.0f};

#pragma unroll 8
    for (int t = 0; t < LIF_T; ++t) {
        v4f x = __builtin_nontemporal_load(ip + (long)t * N4);
        lif_step(x, v, s);
        __builtin_nontemporal_store(s, sp  + (long)t * N4);
        __builtin_nontemporal_store(v, vp  + (long)t * N4);
        __builtin_nontemporal_store(s, sp2 + (long)t * N4);
    }
}

// Generic runtime-stride fallback (any N divisible by 4). Not hit for the
// reference shapes.
__global__ __launch_bounds__(256, 1) void lif_scan_v4(
    const v4f* __restrict__ I,
    v4f* __restrict__ spk,
    v4f* __restrict__ vm,
    v4f* __restrict__ spk2,
    int n4)
{
    const int i = blockIdx.x * blockDim.x + threadIdx.x;
    if (i >= n4) return;

    const v4f* ip  = I    + i;
    v4f*       sp  = spk  + i;
    v4f*       vp  = vm   + i;
    v4f*       sp2 = spk2 + i;

    v4f v = {0.0f, 0.0f, 0.0f, 0.0f};
    v4f s = {0.0f, 0.0f, 0.0f, 0.0f};

    const long stride = (long)n4;

#pragma unroll 8
    for (int t = 0; t < LIF_T; ++t) {
        v4f x = __builtin_nontemporal_load(ip + (long)t * stride);
        lif_step(x, v, s);
        __builtin_nontemporal_store(s, sp  + (long)t * stride);
        __builtin_nontemporal_store(v, vp  + (long)t * stride);
        __builtin_nontemporal_store(s, sp2 + (long)t * stride);
    }
}

// Scalar tail for N % 4 != 0 (not hit for N = 262144).
__global__ void lif_scan_tail(
    const float* __restrict__ I,
    float* __restrict__ spk,
    float* __restrict__ vm,
    float* __restrict__ spk2,
    int n, int start)
{
    const int i = start + blockIdx.x * blockDim.x + threadIdx.x;
    if (i >= n) return;

    float v = 0.0f, s = 0.0f;
    for (int t = 0; t < LIF_T; ++t) {
        const long o = (long)t * n + i;
        float vv = __builtin_fmaf(LIF_BETA, v, I[o]);
        vv = __builtin_fmaf(-LIF_GAMMA, s, vv);
        s = (vv > LIF_THRESH) ? 1.0f : 0.0f;
        v = (vv > LIF_THRESH) ? 0.0f : vv;
        __builtin_nontemporal_store(s, spk  + o);
        __builtin_nontemporal_store(v, vm   + o);
        __builtin_nontemporal_store(s, spk2 + o);
    }
}

extern "C" void kernel_launch(void* const* d_in, const int* in_sizes, int n_in,
                              void* d_out, int out_size, void* d_ws, size_t ws_size,
                              hipStream_t stream)
{
    (void)n_in; (void)d_ws; (void)ws_size; (void)out_size;

    const float* I = (const float*)d_in[0];
    const int TN = in_sizes[0];        // T * N
    const int N  = TN / LIF_T;         // 262144 for the reference shapes

    float* spikes  = (float*)d_out;         // output 0: (T, N)
    float* vmem    = spikes + (long)TN;     // output 1: (T, N)
    float* spikes2 = vmem   + (long)TN;     // output 2: duplicate spikes

    const int n4  = N / 4;
    const int rem = N - n4 * 4;

    constexpr int N4_REF = 262144 / 4;      // 65536: reference problem size

    if (n4 == N4_REF) {
        const int block = 256;              // 8 wave32's per block
        const int grid  = N4_REF / block;   // 256 blocks, 2048 waves
        lif_scan_v4_ref<<<grid, block, 0, stream>>>(
            (const v4f*)I, (v4f*)spikes, (v4f*)vmem, (v4f*)spikes2);
    } else if (n4 > 0) {
        const int block = 256;
        const int grid  = (n4 + block - 1) / block;
        lif_scan_v4<<<grid, block, 0, stream>>>(
            (const v4f*)I, (v4f*)spikes, (v4f*)vmem, (v4f*)spikes2, n4);
    }
    if (rem > 0) {
        lif_scan_tail<<<1, 64, 0, stream>>>(I, spikes, vmem, spikes2, N, n4 * 4);
    }
}